// Counter_29162827939861
// MI455X (gfx1250) — compile-verified
//
#include <hip/hip_runtime.h>
#include <math.h>
#include <stdint.h>

#define N_PLIN 16
#define OBJ    10
#define M_IN   36
#define WAVES  8
#define BLOCK  (WAVES * 32)

typedef uint32_t v4u __attribute__((ext_vector_type(4)));
typedef int      v8i __attribute__((ext_vector_type(8)));
typedef int      v4i __attribute__((ext_vector_type(4)));

#if defined(__has_builtin)
#if __has_builtin(__builtin_amdgcn_tensor_load_to_lds) && __has_builtin(__builtin_amdgcn_s_wait_tensorcnt)
#define USE_TDM 1
#endif
#endif
#ifndef USE_TDM
#define USE_TDM 0
#endif

// PiecewiseLin with precomputed normalized weights (wn) and inclusive cumsum (cs).
__device__ __forceinline__ float plinf(const float* cs, const float* wn, float x) {
    float y  = 16.0f * x;
    int   id = (int)y;              // trunc toward zero, x >= 0
    float f  = y - (float)id;
    int i0 = id < N_PLIN ? id : N_PLIN;
    int i1 = (id + 1) < N_PLIN ? (id + 1) : N_PLIN;
    return cs[i0] + f * wn[i1];
}

__device__ __forceinline__ float wave_sum(float v) {
#pragma unroll
    for (int off = 16; off > 0; off >>= 1) v += __shfl_xor(v, off, 32);
    return v;
}

#if USE_TDM
// Build D# for a 1-D TDM copy of nelem f32 elements (global -> LDS).
// Group0 {count=1 | lds_addr | global_addr | type=2}; group1 {data_size=4B,
// tensor_dim0=nelem, tensor_dim1=1, tile_dim0=nelem, tensor_dim0_stride=nelem}.
__device__ __forceinline__ void tdm_build_1d(uint32_t lds_off, const float* gptr,
                                             uint32_t nelem, v4u* G0, v8i* G1) {
    uint64_t ga = (uint64_t)(uintptr_t)gptr;
    v4u g0;
    g0[0] = 1u;                                      // count=1 (valid user descriptor)
    g0[1] = lds_off;                                 // lds_addr (bytes)
    g0[2] = (uint32_t)ga;                            // global_addr[31:0]
    g0[3] = (uint32_t)(ga >> 32) | 0x80000000u;      // global_addr[56:32] | type=2<<30
    v8i g1;
    g1[0] = 0x20000;                                 // data_size=2 (4 bytes), wg_mask=0
    g1[1] = (int)((nelem & 0xFFFFu) << 16);          // tensor_dim0[15:0]
    g1[2] = (int)((nelem >> 16) | (1u << 16));       // tensor_dim0[31:16] | tensor_dim1=1
    g1[3] = (int)((nelem & 0xFFFFu) << 16);          // tile_dim0 = nelem
    g1[4] = 0;                                       // tile_dim1/2 unused
    g1[5] = (int)nelem;                              // tensor_dim0_stride[31:0]
    g1[6] = 0;
    g1[7] = 0;
    *G0 = g0; *G1 = g1;
}
#endif

__global__ __launch_bounds__(BLOCK) void counter_kernel(
    const float* __restrict__ boxes,   // [n, 4, 36]
    const float* __restrict__ attn,    // [n, 36]
    const float* __restrict__ ws,      // [8, 17]
    float* __restrict__ out,           // [n, 11]
    int n)
{
    __shared__ float wnL[8 * 17];
    __shared__ float csL[8 * 17];
    __shared__ float attSlab[WAVES * M_IN];
    __shared__ float boxSlab[WAVES * 4 * M_IN];
    __shared__ float attS[WAVES][OBJ];
    __shared__ float b0L[WAVES][OBJ], b1L[WAVES][OBJ], b2L[WAVES][OBJ], b3L[WAVES][OBJ];
    __shared__ float arL[WAVES][OBJ];
    __shared__ float dedupL[WAVES][OBJ * OBJ];
    __shared__ float scoreL[WAVES][OBJ * OBJ];
    __shared__ float simL[WAVES][OBJ * OBJ];
    __shared__ float rsL[WAVES][OBJ];

    const int tid  = threadIdx.x;
    const int lane = tid & 31;
    const int wave = tid >> 5;
    const int blockBase = blockIdx.x * WAVES;
    const int sample = blockBase + wave;
    const bool act = sample < n;

    int cnt = n - blockBase;
    if (cnt > WAVES) cnt = WAVES;
    if (cnt < 0) cnt = 0;
    const int elemsA = cnt * M_IN;
    const int elemsB = cnt * 4 * M_IN;

    // ---- Stage this block's inputs into LDS (TDM async DMA) ----
#if USE_TDM
    {
        // Block-uniform descriptor construction (hoisted so it codegens scalar).
        v4u g0b, g0a; v8i g1b, g1a;
        tdm_build_1d((uint32_t)(uintptr_t)&boxSlab[0],
                     boxes + (size_t)blockBase * 4 * M_IN, (uint32_t)elemsB, &g0b, &g1b);
        tdm_build_1d((uint32_t)(uintptr_t)&attSlab[0],
                     attn + (size_t)blockBase * M_IN, (uint32_t)elemsA, &g0a, &g1a);
        v4i z4; z4[0] = z4[1] = z4[2] = z4[3] = 0;
        v8i z8; z8[0] = z8[1] = z8[2] = z8[3] = z8[4] = z8[5] = z8[6] = z8[7] = 0;
        if (wave == 0) {   // one issuing wave per block; TDM is a wave-level op
            __builtin_amdgcn_tensor_load_to_lds(g0b, g1b, z4, z4, z8, 0);
            __builtin_amdgcn_tensor_load_to_lds(g0a, g1a, z4, z4, z8, 0);
        }
    }
#else
    for (int i = tid; i < elemsB; i += BLOCK)
        boxSlab[i] = boxes[(size_t)blockBase * 4 * M_IN + i];
    for (int i = tid; i < elemsA; i += BLOCK)
        attSlab[i] = attn[(size_t)blockBase * M_IN + i];
#endif

    // ---- Normalize ws -> wn = |w|/sum, cs = cumsum(wn); overlaps with DMA ----
    if (tid < 8) {
        const float* w = ws + tid * 17;
        float t[17]; float s = 0.f;
#pragma unroll
        for (int i = 0; i < 17; ++i) { t[i] = fabsf(w[i]); s += t[i]; }
        float inv = 1.f / s, c = 0.f;
#pragma unroll
        for (int i = 0; i < 17; ++i) {
            float v = t[i] * inv;
            wnL[tid * 17 + i] = v;
            c += v;
            csL[tid * 17 + i] = c;
        }
    }

#if USE_TDM
    __builtin_amdgcn_s_wait_tensorcnt(0);   // no-op for waves that issued nothing
#endif
    __syncthreads();

#define PL(r, x) plinf(csL + (r) * 17, wnL + (r) * 17, (x))

    // ---- top-10 of 36 via wave32 xor-butterfly argmax (order-agnostic downstream) ----
    float av = attSlab[wave * M_IN + lane];
    float bv = (lane < (M_IN - 32)) ? attSlab[wave * M_IN + 32 + lane] : -__builtin_inff();
    int myKsel = 0;
#pragma unroll
    for (int k = 0; k < OBJ; ++k) {
        float v; int ci;
        if (av >= bv) { v = av; ci = lane; } else { v = bv; ci = 32 + lane; }
#pragma unroll
        for (int off = 16; off > 0; off >>= 1) {
            float ov = __shfl_xor(v, off, 32);
            int   oi = __shfl_xor(ci, off, 32);
            if (ov > v || (ov == v && oi < ci)) { v = ov; ci = oi; }
        }
        if (lane == k) myKsel = ci;
        if (ci < 32) { if (lane == ci) av = -__builtin_inff(); }
        else         { if (lane == ci - 32) bv = -__builtin_inff(); }
    }

    // ---- gather selected objects, sigmoid(att), box coords, areas ----
    if (lane < OBJ) {
        float a = attSlab[wave * M_IN + myKsel];
        float s = 1.f / (1.f + expf(-a));
        attS[wave][lane] = s;
        const float* B = &boxSlab[wave * 4 * M_IN];
        float x0 = B[0 * M_IN + myKsel];
        float y0 = B[1 * M_IN + myKsel];
        float x1 = B[2 * M_IN + myKsel];
        float y1 = B[3 * M_IN + myKsel];
        b0L[wave][lane] = x0; b1L[wave][lane] = y0;
        b2L[wave][lane] = x1; b3L[wave][lane] = y1;
        arL[wave][lane] = fmaxf(x1 - x0, 0.f) * fmaxf(y1 - y0, 0.f);
    }
    __syncthreads();

    // ---- pass 1: per-pair relev/dist, score, dedup, simA, dist_conf ----
    float dcAcc = 0.f;   // sum |plin6(dist)-0.5| over 100 pairs
#pragma unroll
    for (int t = 0; t < 4; ++t) {
        int p = t * 32 + lane;
        if (p < OBJ * OBJ) {
            int i = p / OBJ, j = p - i * OBJ;
            float si = attS[wave][i], sj = attS[wave][j];
            float relev = si * sj;
            float mnx = fmaxf(b0L[wave][i], b0L[wave][j]);
            float mny = fmaxf(b1L[wave][i], b1L[wave][j]);
            float mxx = fminf(b2L[wave][i], b2L[wave][j]);
            float mxy = fminf(b3L[wave][i], b3L[wave][j]);
            float iw = fmaxf(mxx - mnx, 0.f), ih = fmaxf(mxy - mny, 0.f);
            float inter = iw * ih;
            float iou = inter / (arL[wave][i] + arL[wave][j] - inter + 1e-12f);
            float dist = 1.f - iou;
            scoreL[wave][p] = PL(0, relev) * PL(1, dist);
            dedupL[wave][p] = PL(3, relev) * PL(4, dist);
            simL[wave][p]   = PL(2, 1.f - fabsf(si - sj));
            dcAcc += fabsf(PL(6, dist) - 0.5f);
        }
    }
    __syncthreads();

    // ---- pass 2: sim[i][j] *= prod_a plin2(1 - |dedup[a][i] - dedup[a][j]|)  (o^3) ----
#pragma unroll
    for (int t = 0; t < 4; ++t) {
        int p = t * 32 + lane;
        if (p < OBJ * OBJ) {
            int i = p / OBJ, j = p - i * OBJ;
            float prod = simL[wave][p];
#pragma unroll
            for (int a = 0; a < OBJ; ++a) {
                float d = fabsf(dedupL[wave][a * OBJ + i] - dedupL[wave][a * OBJ + j]);
                prod *= PL(2, 1.f - d);
            }
            simL[wave][p] = prod;
        }
    }
    __syncthreads();

    // ---- row sums of sim ----
    if (lane < OBJ) {
        float rs = 0.f;
#pragma unroll
        for (int j = 0; j < OBJ; ++j) rs += simL[wave][lane * OBJ + j];
        rsL[wave][lane] = rs;
    }
    __syncthreads();

    // ---- totals: sum(score/all_sims) + sum(correction); confidence terms ----
    float corrAcc = 0.f, acAcc = 0.f;
    if (lane < OBJ) {
        float s = attS[wave][lane];
        corrAcc = PL(0, s * s) / rsL[wave][lane];
        acAcc   = fabsf(PL(5, s) - 0.5f);
    }
    float scAcc = 0.f;
#pragma unroll
    for (int t = 0; t < 4; ++t) {
        int p = t * 32 + lane;
        if (p < OBJ * OBJ) {
            int i = p / OBJ, j = p - i * OBJ;
            scAcc += scoreL[wave][p] / (rsL[wave][i] * rsL[wave][j]);
        }
    }
    float tot = wave_sum(scAcc + corrAcc);
    float ac  = wave_sum(acAcc) * (1.f / OBJ);
    float dc  = wave_sum(dcAcc) * (1.f / (OBJ * OBJ));

    float total = sqrtf(tot + 1e-20f);
    float conf  = PL(7, ac + dc);

    // ---- to_one_hot * conf ----
    float s = fminf(fmaxf(total, 0.f), (float)OBJ);
    int   ii = (int)s;
    float f  = s - (float)ii;
    int tl = ii < OBJ ? ii : OBJ;
    int tr = (ii + 1) < OBJ ? (ii + 1) : OBJ;
    if (act && lane < OBJ + 1) {
        float v = 0.f;
        if (lane == tl) v += 1.f - f;
        if (lane == tr) v += f;
        out[(size_t)sample * (OBJ + 1) + lane] = v * conf;
    }
#undef PL
}

extern "C" void kernel_launch(void* const* d_in, const int* in_sizes, int n_in,
                              void* d_out, int out_size, void* d_ws, size_t ws_size,
                              hipStream_t stream) {
    (void)n_in; (void)out_size; (void)d_ws; (void)ws_size;
    const float* boxes = (const float*)d_in[0];   // [n, 4, 36] f32
    const float* attn  = (const float*)d_in[1];   // [n, 36]    f32
    const float* ws    = (const float*)d_in[2];   // [8, 17]    f32
    float* out = (float*)d_out;                   // [n, 11]    f32
    int n = in_sizes[1] / M_IN;
    int blocks = (n + WAVES - 1) / WAVES;
    counter_kernel<<<blocks, BLOCK, 0, stream>>>(boxes, attn, ws, out, n);
}